// QuantumCircuitLayer_24275155157089
// MI455X (gfx1250) — compile-verified
//
#include <hip/hip_runtime.h>
#include <hip/hip_bf16.h>

typedef __attribute__((ext_vector_type(2))) float v2f;
typedef __attribute__((ext_vector_type(8))) float v8f;

// ---------- complex helpers (setup kernel) ----------
__device__ __forceinline__ float2 cmul(float2 a, float2 b) {
    return make_float2(a.x * b.x - a.y * b.y, a.x * b.y + a.y * b.x);
}
__device__ __forceinline__ float2 cadd(float2 a, float2 b) {
    return make_float2(a.x + b.x, a.y + b.y);
}
__device__ __forceinline__ float2 cscale(float2 a, float s) {
    return make_float2(a.x * s, a.y * s);
}
__device__ __forceinline__ float2 cis(float a) {
    float s, c;
    sincosf(a, &s, &c);
    return make_float2(c, s);
}

// ds_swizzle group-of-32 XOR pattern: offset[15]=0, and_mask=0x1F (offset[4:0]),
// or_mask=0 (offset[9:5]), xor_mask=D (offset[14:10])  ->  SWAPX{D}
// Butterfly level: p <- (lane & D) ? swz(p) - p : p + swz(p)   (Walsh-Hadamard)
template <int D>
__device__ __forceinline__ void wht_level(float (&p)[8], int lane) {
    constexpr int OFF = (D << 10) | 0x1F;  // immediate swizzle pattern
#pragma unroll
    for (int v = 0; v < 8; ++v) {
        const float y =
            __int_as_float(__builtin_amdgcn_ds_swizzle(__float_as_int(p[v]), OFF));
        p[v] = (lane & D) ? (y - p[v]) : (p[v] + y);
    }
}

// =====================================================================
// Kernel 1: fold the 6 StronglyEntangling layers (Rot gates + CNOT ring)
// into one 16x16 complex unitary U. 256 threads: thread = (row, col).
// Writes Ur (16x16 row-major) then Ui to d_ws.
// =====================================================================
__global__ __launch_bounds__(256) void build_unitary_kernel(
    const float* __restrict__ qw, float* __restrict__ Ur, float* __restrict__ Ui) {
    __shared__ float2 lds[256];  // [row*16 + col]
    const int t = threadIdx.x;
    const int row = t >> 4;
    const int col = t & 15;

    float2 u = make_float2(row == col ? 1.0f : 0.0f, 0.0f);

    for (int l = 0; l < 6; ++l) {
        // Rot(phi, theta, omega) = RZ(omega) RY(theta) RZ(phi) on each wire
        for (int w = 0; w < 4; ++w) {
            const float* g = qw + (l * 4 + w) * 3;
            const float phi = g[0], th = g[1], om = g[2];
            float st, ct;
            sincosf(0.5f * th, &st, &ct);
            const float2 m00 = cscale(cis(-0.5f * (phi + om)), ct);
            const float2 m11 = cscale(cis(0.5f * (phi + om)), ct);
            const float2 m01 = cscale(cis(0.5f * (phi - om)), -st);
            const float2 m10 = cscale(cis(-0.5f * (phi - om)), st);

            const int mask = 8 >> w;  // wire 0 = MSB of the 4-bit index
            lds[t] = u;
            __syncthreads();
            const float2 up = lds[(row ^ mask) * 16 + col];
            if ((row & mask) == 0)
                u = cadd(cmul(m00, u), cmul(m01, up));
            else
                u = cadd(cmul(m10, up), cmul(m11, u));
            __syncthreads();
        }
        // ring of CNOTs, control w -> target (w+r)%4, applied w = 0..3.
        // Pure row permutation; compose: v'[row] = v[s0(s1(s2(s3(row))))].
        const int r = (l % 3) + 1;
        int src = row;
        for (int w = 3; w >= 0; --w) {
            const int tw = (w + r) & 3;
            if ((src >> (3 - w)) & 1) src ^= (8 >> tw);
        }
        lds[t] = u;
        __syncthreads();
        u = lds[src * 16 + col];
        __syncthreads();
    }
    Ur[row * 16 + col] = u.x;
    Ui[row * 16 + col] = u.y;
}

// =====================================================================
// Kernel 2: per-sample work. One wave (32 lanes) per 16-sample tile.
//   v[j]   = prod_w (bit ? sin(x_w/2) : cos(x_w/2))   (real product state)
//   F_re   = v · Ur^T , F_im = v · Ui^T   via 4+4 chained WMMA f32 16x16x4
//   p      = F_re^2 + F_im^2
//   out[w] = Walsh-Hadamard of p at masks 8,4,2,1 (== <Z_w>)
// =====================================================================
__global__ __launch_bounds__(256) void qsim16_wmma_kernel(
    const float* __restrict__ inp,   // (B,4) angles
    const float* __restrict__ Ur,    // 16x16 row-major
    const float* __restrict__ Ui,    // 16x16 row-major
    float* __restrict__ out,         // (B,4)
    int ntiles) {
    const int lane = threadIdx.x & 31;
    const int wave = threadIdx.x >> 5;
    const int tile = blockIdx.x * 8 + wave;  // 8 waves per block
    if (tile >= ntiles) return;              // wave-uniform guard (EXEC stays all-ones)

    const int n = lane & 15;   // A-row (sample in tile) AND D-column (basis state k)
    const int h = lane >> 4;   // half-wave select: K-halves / sample-halves

    // ---- per-sample embedding amplitudes (rank-1 product state) ----
    const float4 x = reinterpret_cast<const float4*>(inp)[tile * 16 + n];
    float s0, c0, s1, c1, s2, c2, s3, c3;
    __sincosf(0.5f * x.x, &s0, &c0);
    __sincosf(0.5f * x.y, &s1, &c1);
    __sincosf(0.5f * x.z, &s2, &c2);
    __sincosf(0.5f * x.w, &s3, &c3);
    // amp[j] = t01[j>>2] * t23[j&3]
    const float t01[4] = {c0 * c1, c0 * s1, s0 * c1, s0 * s1};
    const float t23[4] = {c2 * c3, c2 * s3, s2 * c3, s2 * s3};
    // A fragment needs amp[4c + 2h + v]  ->  t01[c] * t23[2h+v]
    const float q0 = h ? t23[2] : t23[0];
    const float q1 = h ? t23[3] : t23[1];

    // ---- B fragments: B[Klocal, N] = U[N, 4c + Klocal], Klocal = 2h+v ----
    const float2* Ur2 = reinterpret_cast<const float2*>(Ur);
    const float2* Ui2 = reinterpret_cast<const float2*>(Ui);
    const int bbase = n * 8 + h;  // float2 index of U[n, 4c+2h]

    v8f accR = {0.f, 0.f, 0.f, 0.f, 0.f, 0.f, 0.f, 0.f};
    v8f accI = {0.f, 0.f, 0.f, 0.f, 0.f, 0.f, 0.f, 0.f};

#pragma unroll
    for (int c = 0; c < 4; ++c) {
        const float a = t01[c];
        v2f A;
        A[0] = a * q0;
        A[1] = a * q1;
        const float2 br = Ur2[bbase + 2 * c];
        const float2 bi = Ui2[bbase + 2 * c];
        v2f Br; Br[0] = br.x; Br[1] = br.y;
        v2f Bi; Bi[0] = bi.x; Bi[1] = bi.y;
        accR = __builtin_amdgcn_wmma_f32_16x16x4_f32(false, A, false, Br, (short)0,
                                                     accR, false, false);
        accI = __builtin_amdgcn_wmma_f32_16x16x4_f32(false, A, false, Bi, (short)0,
                                                     accI, false, false);
    }

    // ---- probabilities: lane holds column k=n for samples v+8h (v=0..7) ----
    float p[8];
#pragma unroll
    for (int v = 0; v < 8; ++v)
        p[v] = accR[v] * accR[v] + accI[v] * accI[v];

    // ---- 16-point Walsh-Hadamard over k (lanes) per half-wave, via
    //      DS_SWIZZLE group-of-32 XOR patterns (immediate, no index VGPRs) ----
    wht_level<1>(p, lane);
    wht_level<2>(p, lane);
    wht_level<4>(p, lane);
    wht_level<8>(p, lane);

    // ---- masks 8,4,2,1 are exactly <Z_0..3>; write (B,4) output ----
    const int m = n;
    if (m == 1 || m == 2 || m == 4 || m == 8) {
        const int w = (m == 8) ? 0 : (m == 4) ? 1 : (m == 2) ? 2 : 3;
        const int sbase = tile * 16 + 8 * h;
#pragma unroll
        for (int v = 0; v < 8; ++v)
            out[(sbase + v) * 4 + w] = p[v];
    }
}

extern "C" void kernel_launch(void* const* d_in, const int* in_sizes, int n_in,
                              void* d_out, int out_size, void* d_ws, size_t ws_size,
                              hipStream_t stream) {
    const float* inputs = (const float*)d_in[0];   // (B,4) float32
    const float* qweights = (const float*)d_in[1]; // (6,4,3) float32
    float* out = (float*)d_out;                    // (B,4) float32

    float* Ur = (float*)d_ws;       // 256 floats
    float* Ui = Ur + 256;           // 256 floats (ws >= 2 KB)

    const int B = in_sizes[0] / 4;
    const int ntiles = B / 16;      // B = 262144 -> 16384 tiles

    build_unitary_kernel<<<1, 256, 0, stream>>>(qweights, Ur, Ui);

    const int waves_per_block = 8;
    const int blocks = (ntiles + waves_per_block - 1) / waves_per_block;
    qsim16_wmma_kernel<<<blocks, 256, 0, stream>>>(inputs, Ur, Ui, out, ntiles);
}